// BahdanauAttention_70720931496613
// MI455X (gfx1250) — compile-verified
//
#include <hip/hip_runtime.h>
#include <math.h>

// ---------------------------------------------------------------------------
// Bahdanau attention, MI455X (gfx1250, wave32, WMMA + async Global->LDS).
//   features [B,L,D] fp32, hidden [B,H], W1 [D,U], W2 [H,U], V [U,1]
//   B=128 L=2048 D=256 H=512 U=256
// Memory-bound: 2 streaming passes over 256MB features (~22us @ 23.3TB/s).
// GEMM (34.4 GFLOP fp32) runs on V_WMMA_F32_16X16X4_F32 (fp32 fidelity).
// LDS staging uses GLOBAL_LOAD_ASYNC_TO_LDS_B128 (ASYNCcnt) -> no VGPR
// round-trip, single s_wait_asynccnt before each barrier.
// ---------------------------------------------------------------------------

typedef float v2f __attribute__((ext_vector_type(2)));
typedef float v8f __attribute__((ext_vector_type(8)));

#define B_ 128
#define L_ 2048
#define D_ 256
#define H_ 512
#define U_ 256

// workspace layout (float offsets)
#define WS_WT     0                      // 16*64*64 = 65536  (W1 pre-shuffled for B-operand)
#define WS_PH     (WS_WT + 65536)        // B*U = 32768       (hidden@W2 + b1 + b2)
#define WS_LOGITS (WS_PH + (B_ * U_))    // B*L = 262144
#define WS_CTXP   (WS_LOGITS + (B_ * L_))// B*8*D = 262144

// ---------------------------------------------------------------------------
// gfx1250 async Global->LDS copy, GVS addressing:
//   each lane copies 16 bytes from global address sbase plus g_byte_off to LDS offset lds_byte_off.
// Dynamic LDS starts at offset 0 for this kernel (no static __shared__), so
// the VDST VGPR is simply the byte offset into our smem layout.
// ---------------------------------------------------------------------------
__device__ __forceinline__ void async_copy_b128(unsigned lds_byte_off,
                                                const void* sbase,
                                                int g_byte_off) {
  asm volatile("global_load_async_to_lds_b128 %0, %1, %2"
               :: "v"(lds_byte_off), "v"(g_byte_off), "s"(sbase)
               : "memory");
}
__device__ __forceinline__ void wait_async0() {
  asm volatile("s_wait_asynccnt 0x0" ::: "memory");
}

// ---------------------------------------------------------------------------
// Pre-shuffle W1 [D,U] into WMMA B-operand order:
// for (ut, ks): lane n<16  -> {W1[4ks+0][ut*16+n], W1[4ks+1][ut*16+n]}
//               lane n+16  -> {W1[4ks+2][...],     W1[4ks+3][...]}
// linear: Wt[((ut*64+ks)*32 + lane)*2 + j]
// ---------------------------------------------------------------------------
__global__ __launch_bounds__(256) void k_prep_wt(const float* __restrict__ W1,
                                                 float* __restrict__ Wt) {
  int tid  = blockIdx.x * 256 + threadIdx.x;   // 0..32767
  int lane = tid & 31;
  int ks   = (tid >> 5) & 63;
  int ut   = tid >> 11;
  int n    = ut * 16 + (lane & 15);
  int k    = ks * 4 + ((lane >> 4) << 1);
  v2f v;
  v.x = W1[k * U_ + n];
  v.y = W1[(k + 1) * U_ + n];
  *(v2f*)(Wt + (size_t)((ut * 64 + ks) * 32 + lane) * 2) = v;
}

// ---------------------------------------------------------------------------
// ph[b][u] = hidden[b]·W2[:,u] + b1[u] + b2[u]   (bV cancels in softmax)
// ---------------------------------------------------------------------------
__global__ __launch_bounds__(256) void k_proj_h(const float* __restrict__ hidden,
                                                const float* __restrict__ W2,
                                                const float* __restrict__ b1,
                                                const float* __restrict__ b2,
                                                float* __restrict__ ph) {
  int b = blockIdx.x, u = threadIdx.x;
  float acc = b1[u] + b2[u];
  const float* hb = hidden + b * H_;
  for (int h = 0; h < H_; ++h) acc = fmaf(hb[h], W2[h * U_ + u], acc);
  ph[b * U_ + u] = acc;
}

// ---------------------------------------------------------------------------
// Main fused kernel: logits[b][l] = sum_u tanh( (features@W1)[l,u] + ph[b,u] ) * V[u]
// Grid (B, L/128), block 256 (8 waves). Wave w owns rows [l0+16w, l0+16w+16).
// LDS: sF[128][260] (pad 4 -> conflict-free ds_load_b64 A reads, 16B-aligned
//      async b128 stores: 260*4 = 1040 = 65*16), sB[64][64] staged per u-tile.
// A panel lives in 128 VGPRs/lane; all staging is async Global->LDS.
// ---------------------------------------------------------------------------
#define SB_FLOAT_OFF (128 * 260)   // sB starts here in smem (floats)

__global__ __launch_bounds__(256) void k_logits(const float* __restrict__ feat,
                                                const float* __restrict__ Wt,
                                                const float* __restrict__ ph,
                                                const float* __restrict__ V,
                                                float* __restrict__ logits) {
  extern __shared__ float smem[];
  float* sF = smem;                    // 128 * 260 = 33280 floats

  const int t    = threadIdx.x;
  const int lane = t & 31;
  const int wave = t >> 5;
  const int b    = blockIdx.x;
  const int l0   = blockIdx.y * 128;

  // ---- stage features tile [128 x 256] via async b128, coalesced ----
  const float* fbase = feat + ((size_t)(b * L_ + l0)) * D_;
#pragma unroll
  for (int i = 0; i < 32; ++i) {
    int idx4 = i * 256 + t;            // 8192 float4 slots
    int r = idx4 >> 6, c4 = idx4 & 63;
    unsigned lds_off = (unsigned)(r * 260 + c4 * 4) * 4u;   // bytes, 16B aligned
    int g_off = (r * D_ + c4 * 4) * 4;                      // bytes
    async_copy_b128(lds_off, fbase, g_off);
  }
  wait_async0();
  __syncthreads();

  // ---- A-operand panel into registers: lane m=lane&15, K pair at 4ks+2*(lane>>4) ----
  v2f a[64];
  {
    const int rowl = lane & 15;
    const int koff = (lane >> 4) * 2;
    const float* aptr = sF + (wave * 16 + rowl) * 260 + koff;
#pragma unroll
    for (int ks = 0; ks < 64; ++ks) a[ks] = *(const v2f*)(aptr + ks * 4);
  }

  float logacc[8] = {0.f, 0.f, 0.f, 0.f, 0.f, 0.f, 0.f, 0.f};
  const unsigned sb_byte = (unsigned)SB_FLOAT_OFF * 4u;
  const float* sBr = smem + SB_FLOAT_OFF;

  for (int ut = 0; ut < 16; ++ut) {
    __syncthreads();  // previous sB reads done
    // stage B tile (4096 floats) via async b128, coalesced
#pragma unroll
    for (int j = 0; j < 4; ++j) {
      unsigned lds_off = sb_byte + (unsigned)(j * 1024 + t * 4) * 4u;
      int g_off = (ut * 4096 + j * 1024 + t * 4) * 4;
      async_copy_b128(lds_off, Wt, g_off);
    }
    wait_async0();
    __syncthreads();

    v8f acc = {};
#pragma unroll
    for (int ks = 0; ks < 64; ++ks) {
      v2f bv = *(const v2f*)(sBr + ks * 64 + lane * 2);
      acc = __builtin_amdgcn_wmma_f32_16x16x4_f32(
          /*neg_a=*/false, a[ks], /*neg_b=*/false, bv,
          /*c_mod=*/(short)0, acc, /*reuse_a=*/false, /*reuse_b=*/false);
    }
    // C layout: lane holds N = lane&15 (+16*ut); VGPR i -> M = (lane>>4)*8 + i
    float phu = ph[b * U_ + ut * 16 + (lane & 15)];
    float vu  = V[ut * 16 + (lane & 15)];
#pragma unroll
    for (int i = 0; i < 8; ++i) logacc[i] += tanhf(acc[i] + phu) * vu;
  }

  // reduce over the 16 N-lanes within each half-wave
#pragma unroll
  for (int i = 0; i < 8; ++i) {
    float v = logacc[i];
    v += __shfl_xor(v, 1);
    v += __shfl_xor(v, 2);
    v += __shfl_xor(v, 4);
    v += __shfl_xor(v, 8);
    logacc[i] = v;
  }
  if ((lane & 15) == 0) {
    int mbase = l0 + wave * 16 + (lane >> 4) * 8;
#pragma unroll
    for (int i = 0; i < 8; ++i) logits[(size_t)b * L_ + mbase + i] = logacc[i];
  }
}

// ---------------------------------------------------------------------------
// softmax over L per batch -> attention weights (written to d_out tail)
// ---------------------------------------------------------------------------
__global__ __launch_bounds__(256) void k_softmax(const float* __restrict__ logits,
                                                 float* __restrict__ wout) {
  __shared__ float red[256];
  int b = blockIdx.x, t = threadIdx.x;
  const float* lg = logits + (size_t)b * L_;

  float mx = -3.0e38f;
  for (int i = t; i < L_; i += 256) mx = fmaxf(mx, lg[i]);
  red[t] = mx;
  __syncthreads();
  for (int s = 128; s > 0; s >>= 1) {
    if (t < s) red[t] = fmaxf(red[t], red[t + s]);
    __syncthreads();
  }
  mx = red[0];
  __syncthreads();

  float sum = 0.f;
  for (int i = t; i < L_; i += 256) sum += expf(lg[i] - mx);
  red[t] = sum;
  __syncthreads();
  for (int s = 128; s > 0; s >>= 1) {
    if (t < s) red[t] += red[t + s];
    __syncthreads();
  }
  float inv = 1.0f / red[0];
  for (int i = t; i < L_; i += 256)
    wout[(size_t)b * L_ + i] = expf(lg[i] - mx) * inv;
}

// ---------------------------------------------------------------------------
// context partials: grid (B, 8); thread d sums 256 rows (deterministic, no atomics)
// ---------------------------------------------------------------------------
__global__ __launch_bounds__(256) void k_ctx_part(const float* __restrict__ feat,
                                                  const float* __restrict__ w,
                                                  float* __restrict__ part) {
  int b = blockIdx.x, chunk = blockIdx.y, d = threadIdx.x;
  const float* fb = feat + ((size_t)b * L_ + chunk * 256) * D_ + d;
  const float* wb = w + (size_t)b * L_ + chunk * 256;
  float acc = 0.f;
  for (int l = 0; l < 256; ++l) acc = fmaf(wb[l], fb[(size_t)l * D_], acc);
  part[(size_t)(b * 8 + chunk) * D_ + d] = acc;
}

__global__ __launch_bounds__(256) void k_ctx_reduce(const float* __restrict__ part,
                                                    float* __restrict__ ctx) {
  int b = blockIdx.x, d = threadIdx.x;
  float acc = 0.f;
#pragma unroll
  for (int c = 0; c < 8; ++c) acc += part[(size_t)(b * 8 + c) * D_ + d];
  ctx[(size_t)b * D_ + d] = acc;
}

// ---------------------------------------------------------------------------
extern "C" void kernel_launch(void* const* d_in, const int* in_sizes, int n_in,
                              void* d_out, int out_size, void* d_ws, size_t ws_size,
                              hipStream_t stream) {
  const float* features = (const float*)d_in[0];
  const float* hidden   = (const float*)d_in[1];
  const float* W1       = (const float*)d_in[2];
  const float* b1       = (const float*)d_in[3];
  const float* W2       = (const float*)d_in[4];
  const float* b2       = (const float*)d_in[5];
  const float* V        = (const float*)d_in[6];
  // d_in[7] = bV: scalar shift of all logits, cancels in softmax.

  float* ws      = (float*)d_ws;
  float* Wt      = ws + WS_WT;
  float* ph      = ws + WS_PH;
  float* logits  = ws + WS_LOGITS;
  float* ctxp    = ws + WS_CTXP;

  float* ctx_out = (float*)d_out;               // [B, D]
  float* w_out   = (float*)d_out + B_ * D_;     // [B, L, 1]

  k_prep_wt<<<128, 256, 0, stream>>>(W1, Wt);
  k_proj_h<<<B_, U_, 0, stream>>>(hidden, W2, b1, b2, ph);

  size_t shmem = (size_t)(128 * 260 + 64 * 64) * sizeof(float);  // 149504 B
  k_logits<<<dim3(B_, L_ / 128), 256, shmem, stream>>>(features, Wt, ph, V, logits);

  k_softmax<<<B_, 256, 0, stream>>>(logits, w_out);
  k_ctx_part<<<dim3(B_, 8), 256, 0, stream>>>(features, w_out, ctxp);
  k_ctx_reduce<<<B_, 256, 0, stream>>>(ctxp, ctx_out);
}